// DGA_89601607729378
// MI455X (gfx1250) — compile-verified
//
#include <hip/hip_runtime.h>

#define N_NODES 100000
#define N_EDGES 1600000
#define IN_F 64
#define HID 32
#define OUT_F 40

typedef __attribute__((ext_vector_type(2))) float v2f;
typedef __attribute__((ext_vector_type(8))) float v8f;

// workspace layout (float offsets)
#define WS_CNT   0
#define WS_AGG1  (N_NODES)                       // N x 64
#define WS_H1    (WS_AGG1 + N_NODES * IN_F)      // N x 32
#define WS_AGG2  (WS_H1 + N_NODES * HID)         // N x 32
#define WS_TOTAL (WS_AGG2 + N_NODES * HID)       // 12,900,000 floats (~51.6 MB)

// ---------------- zero workspace ----------------
__global__ void gnn_zero(float4* p, int n4) {
    int i = blockIdx.x * blockDim.x + threadIdx.x;
    if (i < n4) p[i] = make_float4(0.f, 0.f, 0.f, 0.f);
}

// ---------------- degree count ----------------
__global__ void gnn_count(const int* __restrict__ ei, float* __restrict__ cnt) {
    int e = blockIdx.x * blockDim.x + threadIdx.x;
    if (e < N_EDGES) {
        int d = ei[N_EDGES + e];
        unsafeAtomicAdd(&cnt[d], 1.0f);
    }
}

__global__ void gnn_invdeg(float* __restrict__ cnt) {
    int i = blockIdx.x * blockDim.x + threadIdx.x;
    if (i < N_NODES) cnt[i] = 1.0f / fmaxf(cnt[i], 1.0f);
}

// ---------------- scatter-add feats[src] into agg[dst] ----------------
// F/4 threads cooperate on one edge; float4 gather, fp32 atomics (L2-resident).
template <int F>
__global__ void gnn_scatter(const float* __restrict__ feats,
                            const int* __restrict__ ei,
                            float* __restrict__ agg) {
    const int C = F / 4;
    int id = blockIdx.x * blockDim.x + threadIdx.x;
    if (id >= N_EDGES * C) return;
    int e = id / C;
    int c = id % C;
    int s = ei[e];
    int d = ei[N_EDGES + e];
    float4 v = *(const float4*)(feats + (long long)s * F + c * 4);
    float* o = agg + (long long)d * F + c * 4;
    unsafeAtomicAdd(o + 0, v.x);
    unsafeAtomicAdd(o + 1, v.y);
    unsafeAtomicAdd(o + 2, v.z);
    unsafeAtomicAdd(o + 3, v.w);
}

// ---------------- layer 1: fused dual-GEMM SAGE (WMMA fp32) ----------------
// out[16x32 tile] = relu( (agg*invdeg) @ Wn + Aself @ Ws + (bn+bs) )
template <int K>
__global__ void gnn_sage_gemm(const float* __restrict__ Aagg,
                              const float* __restrict__ invdeg,
                              const float* __restrict__ Aself,
                              const float* __restrict__ Wn, const float* __restrict__ bn,
                              const float* __restrict__ Ws, const float* __restrict__ bs,
                              float* __restrict__ out) {
    int wv   = (blockIdx.x * blockDim.x + threadIdx.x) >> 5;  // wave-uniform
    int lane = threadIdx.x & 31;
    if (wv >= N_NODES / 16) return;   // uniform per wave -> EXEC all-ones at WMMA
    int row0 = wv * 16;
    int am = lane & 15;               // A: row within tile
    int ak = (lane >> 4) << 1;        // A: K sub-offset (lanes 16-31 hold K+2,K+3)
    int gm = row0 + am;
    int n  = lane & 15;               // B/D: column within 16-wide tile
    float sc = invdeg[gm];
    const float* arow = Aagg  + (long long)gm * K;
    const float* srow = Aself + (long long)gm * K;

    v8f acc0 = {};
    v8f acc1 = {};
    #pragma unroll
    for (int k0 = 0; k0 < K; k0 += 4) {
        int kb = k0 + ak;
        v2f a;  a.x = arow[kb] * sc;          a.y = arow[kb + 1] * sc;
        v2f b0; b0.x = Wn[kb * HID + n];      b0.y = Wn[(kb + 1) * HID + n];
        v2f b1; b1.x = Wn[kb * HID + 16 + n]; b1.y = Wn[(kb + 1) * HID + 16 + n];
        acc0 = __builtin_amdgcn_wmma_f32_16x16x4_f32(false, a, false, b0, (short)0, acc0, false, false);
        acc1 = __builtin_amdgcn_wmma_f32_16x16x4_f32(false, a, false, b1, (short)0, acc1, false, false);
    }
    #pragma unroll
    for (int k0 = 0; k0 < K; k0 += 4) {
        int kb = k0 + ak;
        v2f a;  a.x = srow[kb];               a.y = srow[kb + 1];
        v2f b0; b0.x = Ws[kb * HID + n];      b0.y = Ws[(kb + 1) * HID + n];
        v2f b1; b1.x = Ws[kb * HID + 16 + n]; b1.y = Ws[(kb + 1) * HID + 16 + n];
        acc0 = __builtin_amdgcn_wmma_f32_16x16x4_f32(false, a, false, b0, (short)0, acc0, false, false);
        acc1 = __builtin_amdgcn_wmma_f32_16x16x4_f32(false, a, false, b1, (short)0, acc1, false, false);
    }

    int half = lane >> 4;                 // D: vgpr r holds row r + 8*half
    float bias0 = bn[n] + bs[n];
    float bias1 = bn[16 + n] + bs[16 + n];
    #pragma unroll
    for (int r = 0; r < 8; ++r) {
        float v0 = fmaxf(acc0[r] + bias0, 0.0f);
        float v1 = fmaxf(acc1[r] + bias1, 0.0f);
        int m = row0 + half * 8 + r;
        out[(long long)m * HID + n]      = v0;
        out[(long long)m * HID + 16 + n] = v1;
    }
}

// ---------------- layer 2 + logits, fused (WMMA fp32 + LDS handoff) ----------------
// x32 = relu( (agg2*invdeg) @ Wn2 + h1 @ Ws2 + b )  -> d_out[0 : N*32)
// logits = x32 @ Wo + bo                            -> d_out[N*32 : N*32 + N*40)
// The 16x32 x32 tile is handed from C-layout to A-layout through a per-wave
// LDS tile; intra-wave ordering enforced with s_wait_dscnt 0 (no barrier, so
// early-returning tail waves are safe).
__global__ void gnn_layer2_fused(const float* __restrict__ Aagg,
                                 const float* __restrict__ invdeg,
                                 const float* __restrict__ Aself,
                                 const float* __restrict__ Wn, const float* __restrict__ bn,
                                 const float* __restrict__ Ws, const float* __restrict__ bs,
                                 const float* __restrict__ Wo, const float* __restrict__ bo,
                                 float* __restrict__ x32,
                                 float* __restrict__ logits) {
    __shared__ float tile[8][16][HID + 1];   // +1 pad: stride 33 -> conflict-free rows

    int wvb  = threadIdx.x >> 5;                               // wave within block
    int wv   = (blockIdx.x * blockDim.x + threadIdx.x) >> 5;   // wave-uniform
    int lane = threadIdx.x & 31;
    if (wv >= N_NODES / 16) return;
    int row0 = wv * 16;
    int am = lane & 15;
    int ak = (lane >> 4) << 1;
    int gm = row0 + am;
    int n  = lane & 15;
    float sc = invdeg[gm];
    const float* arow = Aagg  + (long long)gm * HID;
    const float* srow = Aself + (long long)gm * HID;

    // ---- SAGE layer 2 ----
    v8f acc0 = {};
    v8f acc1 = {};
    #pragma unroll
    for (int k0 = 0; k0 < HID; k0 += 4) {
        int kb = k0 + ak;
        v2f a;  a.x = arow[kb] * sc;          a.y = arow[kb + 1] * sc;
        v2f b0; b0.x = Wn[kb * HID + n];      b0.y = Wn[(kb + 1) * HID + n];
        v2f b1; b1.x = Wn[kb * HID + 16 + n]; b1.y = Wn[(kb + 1) * HID + 16 + n];
        acc0 = __builtin_amdgcn_wmma_f32_16x16x4_f32(false, a, false, b0, (short)0, acc0, false, false);
        acc1 = __builtin_amdgcn_wmma_f32_16x16x4_f32(false, a, false, b1, (short)0, acc1, false, false);
    }
    #pragma unroll
    for (int k0 = 0; k0 < HID; k0 += 4) {
        int kb = k0 + ak;
        v2f a;  a.x = srow[kb];               a.y = srow[kb + 1];
        v2f b0; b0.x = Ws[kb * HID + n];      b0.y = Ws[(kb + 1) * HID + n];
        v2f b1; b1.x = Ws[kb * HID + 16 + n]; b1.y = Ws[(kb + 1) * HID + 16 + n];
        acc0 = __builtin_amdgcn_wmma_f32_16x16x4_f32(false, a, false, b0, (short)0, acc0, false, false);
        acc1 = __builtin_amdgcn_wmma_f32_16x16x4_f32(false, a, false, b1, (short)0, acc1, false, false);
    }

    // ---- epilogue: bias + relu; write x32 to global AND to per-wave LDS tile ----
    int half = lane >> 4;
    float bias0 = bn[n] + bs[n];
    float bias1 = bn[16 + n] + bs[16 + n];
    #pragma unroll
    for (int r = 0; r < 8; ++r) {
        float v0 = fmaxf(acc0[r] + bias0, 0.0f);
        float v1 = fmaxf(acc1[r] + bias1, 0.0f);
        int mr = half * 8 + r;
        int m  = row0 + mr;
        x32[(long long)m * HID + n]      = v0;
        x32[(long long)m * HID + 16 + n] = v1;
        tile[wvb][mr][n]      = v0;
        tile[wvb][mr][16 + n] = v1;
    }
    // Intra-wave LDS ordering: DS ops of one wave complete in order; wait
    // for the stores, then every lane may read any lane's data.
    __asm__ volatile("s_wait_dscnt 0x0" ::: "memory");

    // ---- logits GEMM: A = LDS tile (16x32), B = Wo (32x40), 3 col tiles ----
    v8f oacc[3] = {{}, {}, {}};
    #pragma unroll
    for (int k0 = 0; k0 < HID; k0 += 4) {
        int kb = k0 + ak;
        v2f a; a.x = tile[wvb][am][kb]; a.y = tile[wvb][am][kb + 1];
        #pragma unroll
        for (int t = 0; t < 3; ++t) {
            int col  = t * 16 + n;
            int colc = col < OUT_F ? col : OUT_F - 1;   // clamped addr: no exec churn
            float msk = col < OUT_F ? 1.0f : 0.0f;
            v2f bb;
            bb.x = Wo[kb * OUT_F + colc] * msk;
            bb.y = Wo[(kb + 1) * OUT_F + colc] * msk;
            oacc[t] = __builtin_amdgcn_wmma_f32_16x16x4_f32(false, a, false, bb, (short)0, oacc[t], false, false);
        }
    }
    #pragma unroll
    for (int t = 0; t < 3; ++t) {
        int col = t * 16 + n;
        if (col < OUT_F) {
            float bias = bo[col];
            #pragma unroll
            for (int r = 0; r < 8; ++r) {
                int m = row0 + half * 8 + r;
                logits[(long long)m * OUT_F + col] = oacc[t][r] + bias;
            }
        }
    }
}

extern "C" void kernel_launch(void* const* d_in, const int* in_sizes, int n_in,
                              void* d_out, int out_size, void* d_ws, size_t ws_size,
                              hipStream_t stream) {
    const float* x       = (const float*)d_in[0];
    const int*   ei      = (const int*)d_in[1];     // [2, E]: row0 = src, row1 = dst
    const float* Wself1  = (const float*)d_in[2];
    const float* bself1  = (const float*)d_in[3];
    const float* Wneigh1 = (const float*)d_in[4];
    const float* bneigh1 = (const float*)d_in[5];
    const float* Wself2  = (const float*)d_in[6];
    const float* bself2  = (const float*)d_in[7];
    const float* Wneigh2 = (const float*)d_in[8];
    const float* bneigh2 = (const float*)d_in[9];
    const float* Wout    = (const float*)d_in[10];
    const float* bout    = (const float*)d_in[11];
    // d_in[12]/d_in[13] (W_lin1/b_lin1) feed a dead value in the reference -> skipped.

    float* ws   = (float*)d_ws;
    float* cnt  = ws + WS_CNT;    // degree -> invdeg (in place)
    float* agg1 = ws + WS_AGG1;
    float* h1   = ws + WS_H1;
    float* agg2 = ws + WS_AGG2;

    float* x32    = (float*)d_out;                            // [N, 32]
    float* logits = (float*)d_out + (size_t)N_NODES * HID;    // [N, 40]

    const int T = 256;
    const int gemmBlocks = ((N_NODES / 16) * 32 + T - 1) / T; // 6250 waves

    gnn_zero<<<(WS_TOTAL / 4 + T - 1) / T, T, 0, stream>>>((float4*)ws, WS_TOTAL / 4);
    gnn_count<<<(N_EDGES + T - 1) / T, T, 0, stream>>>(ei, cnt);
    gnn_invdeg<<<(N_NODES + T - 1) / T, T, 0, stream>>>(cnt);

    // layer 1
    gnn_scatter<IN_F><<<(N_EDGES * (IN_F / 4) + T - 1) / T, T, 0, stream>>>(x, ei, agg1);
    gnn_sage_gemm<IN_F><<<gemmBlocks, T, 0, stream>>>(
        agg1, cnt, x, Wneigh1, bneigh1, Wself1, bself1, h1);

    // layer 2 + logits (fused)
    gnn_scatter<HID><<<(N_EDGES * (HID / 4) + T - 1) / T, T, 0, stream>>>(h1, ei, agg2);
    gnn_layer2_fused<<<gemmBlocks, T, 0, stream>>>(
        agg2, cnt, h1, Wneigh2, bneigh2, Wself2, bself2, Wout, bout, x32, logits);
}